// GINClassifier_29643864277190
// MI455X (gfx1250) — compile-verified
//
#include <hip/hip_runtime.h>
#include <hip/hip_bf16.h>

typedef __attribute__((ext_vector_type(2))) float v2f;
typedef __attribute__((ext_vector_type(8))) float v8f;

#define BN_EPS 1e-5f

__device__ __forceinline__ v8f wmma_f32(v2f a, v2f b, v8f c) {
  // D = A(16x4,f32) * B(4x16,f32) + C(16x16,f32)  -> v_wmma_f32_16x16x4_f32
  return __builtin_amdgcn_wmma_f32_16x16x4_f32(false, a, false, b, (short)0, c, false, false);
}

// Async global->LDS copy of 16 bytes per lane (ASYNCcnt-tracked).
// Low 32 bits of a generic LDS pointer == LDS byte offset (shared aperture).
__device__ __forceinline__ void async_g2l_b128(const float* g, float* l) {
  unsigned loff = (unsigned)(unsigned long long)l;
  asm volatile("global_load_async_to_lds_b128 %0, %1, off"
               :: "v"(loff), "v"(g) : "memory");
}
__device__ __forceinline__ void wait_async0() {
  asm volatile("s_wait_asynccnt 0" ::: "memory");
}

// ---------------------------------------------------------------------------
// Edge aggregation: agg[dst] += feat[src]   (agg pre-initialized with feat/x)
// ---------------------------------------------------------------------------
__global__ void scatter_add10(const float* __restrict__ feat,
                              const int* __restrict__ src,
                              const int* __restrict__ dst,
                              float* __restrict__ agg, int E) {
  int e = blockIdx.x * blockDim.x + threadIdx.x;
  if (e >= E) return;
  long s = (long)src[e] * 10;
  long d = (long)dst[e] * 10;
#pragma unroll
  for (int c = 0; c < 10; ++c)
    atomicAdd(&agg[d + c], feat[s + c]);
}

__global__ void scatter_add64(const float* __restrict__ feat,
                              const int* __restrict__ src,
                              const int* __restrict__ dst,
                              float* __restrict__ agg, int E) {
  long tid = (long)blockIdx.x * blockDim.x + threadIdx.x;
  int e = (int)(tid >> 6);
  int c = (int)(tid & 63);
  if (e >= E) return;
  // src[e]/dst[e] uniform across the wave-half -> coalesced 64-wide row access
  atomicAdd(&agg[(long)dst[e] * 64 + c], feat[(long)src[e] * 64 + c]);
}

// ---------------------------------------------------------------------------
// GIN MLP: hout = relu( relu(BN(hin@W1 + b1)) @ W2 + b2 ), hidden = 64
// Block: 256 threads = 8 waves; 32 nodes per block.
// Wave (mt,nt) computes the 16x16 tile (rows mt*16.., cols nt*16..).
// Weights staged K-pair-interleaved so each B fragment is one ds_load_b64.
// ---------------------------------------------------------------------------
template <int KIN, int KPAD, int SIN>
__global__ void gin_mlp(const float* __restrict__ hin,   // [N, KIN] = x + agg
                        const float* __restrict__ w1,    // [KIN, 64]
                        const float* __restrict__ b1,    // [64]
                        const float* __restrict__ gamma,
                        const float* __restrict__ beta,
                        const float* __restrict__ rmean,
                        const float* __restrict__ rvar,
                        const float* __restrict__ w2,    // [64, 64]
                        const float* __restrict__ b2,    // [64]
                        float* __restrict__ hout,        // [N, 64]
                        int N) {
  constexpr int SWP = 80;  // pair-row stride in v2f (160 dwords == 32 mod 64)
  constexpr int SH = 68;   // activation LDS row stride (== 4 mod 64 banks)
  __shared__ float sIn[32 * SIN];
  __shared__ float sH1[32 * SH];
  __shared__ v2f sW1p[(KPAD / 2) * SWP];
  __shared__ v2f sW2p[32 * SWP];

  const int t = threadIdx.x;
  const int wave = t >> 5;
  const int lane = t & 31;
  const int half = lane >> 4;
  const int idx = lane & 15;
  const int m0 = (wave >> 2) * 16;  // M tile 0..1
  const int n0 = (wave & 3) * 16;   // N tile 0..3
  const int node0 = blockIdx.x * 32;

  // --- kick off async staging of the input tile (KIN==64 fast path) ---
  if (KIN == 64) {
    for (int i = t; i < 32 * 16; i += 256) {
      int r = i >> 4, q = (i & 15) * 4;
      int node = node0 + r;
      float* ldst = &sIn[r * SIN + q];
      if (node < N) {
        async_g2l_b128(hin + (long)node * 64 + q, ldst);
      } else {
        ldst[0] = 0.f; ldst[1] = 0.f; ldst[2] = 0.f; ldst[3] = 0.f;
      }
    }
  } else {
    for (int i = t; i < 32 * KPAD; i += 256) {
      int r = i / KPAD, c = i % KPAD;
      int node = node0 + r;
      sIn[r * SIN + c] =
          (node < N && c < KIN) ? hin[(long)node * KIN + c] : 0.0f;
    }
  }

  // --- stage weights, K-pair interleaved (zero-pad rows beyond KIN) ---
  for (int i = t; i < (KPAD / 2) * 64; i += 256) {
    int pr = i >> 6, n = i & 63;
    v2f w;
    w.x = (2 * pr < KIN) ? w1[(2 * pr) * 64 + n] : 0.0f;
    w.y = (2 * pr + 1 < KIN) ? w1[(2 * pr + 1) * 64 + n] : 0.0f;
    sW1p[pr * SWP + n] = w;
  }
  for (int i = t; i < 32 * 64; i += 256) {
    int pr = i >> 6, n = i & 63;
    v2f w;
    w.x = w2[(2 * pr) * 64 + n];
    w.y = w2[(2 * pr + 1) * 64 + n];
    sW2p[pr * SWP + n] = w;
  }
  if (KIN == 64) wait_async0();
  __syncthreads();

  // --- GEMM1: [16xKPAD] @ [KPADx16] via f32 WMMA, K-step 4 ---
  v8f acc = {};
#pragma unroll
  for (int k = 0; k < KPAD; k += 4) {
    v2f a, b;
    a.x = sIn[(m0 + idx) * SIN + k + 2 * half];
    a.y = sIn[(m0 + idx) * SIN + k + 2 * half + 1];
    b = sW1p[((k >> 1) + half) * SWP + n0 + idx];
    acc = wmma_f32(a, b, acc);
  }

  // --- epilogue 1: bias + BN(eval) + relu -> sH1 ---
  {
    int ch = n0 + idx;
    float scale = gamma[ch] * rsqrtf(rvar[ch] + BN_EPS);
    float shift = beta[ch] + (b1[ch] - rmean[ch]) * scale;
#pragma unroll
    for (int v = 0; v < 8; ++v) {
      int row = m0 + v + 8 * half;
      sH1[row * SH + ch] = fmaxf(acc[v] * scale + shift, 0.0f);
    }
  }
  __syncthreads();

  // --- GEMM2: [16x64] @ [64x16] ---
  v8f acc2 = {};
#pragma unroll
  for (int k = 0; k < 64; k += 4) {
    v2f a, b;
    a.x = sH1[(m0 + idx) * SH + k + 2 * half];
    a.y = sH1[(m0 + idx) * SH + k + 2 * half + 1];
    b = sW2p[((k >> 1) + half) * SWP + n0 + idx];
    acc2 = wmma_f32(a, b, acc2);
  }

  // --- epilogue 2: bias + relu -> global ---
  {
    int ch = n0 + idx;
    float bias = b2[ch];
#pragma unroll
    for (int v = 0; v < 8; ++v) {
      int node = node0 + m0 + v + 8 * half;
      if (node < N)
        hout[(long)node * 64 + ch] = fmaxf(acc2[v] + bias, 0.0f);
    }
  }
}

// ---------------------------------------------------------------------------
// Per-graph pooling accumulators (h >= 0 post-ReLU, so uint atomicMax works)
// ---------------------------------------------------------------------------
__global__ void pool_accum(const float* __restrict__ h,
                           const int* __restrict__ batch,
                           float* __restrict__ sum,
                           unsigned* __restrict__ mx,
                           float* __restrict__ cnt, int N) {
  long tid = (long)blockIdx.x * blockDim.x + threadIdx.x;
  int node = (int)(tid >> 6);
  int c = (int)(tid & 63);
  if (node >= N) return;
  int g = batch[node];
  float v = h[(long)node * 64 + c];
  atomicAdd(&sum[(long)g * 64 + c], v);
  atomicMax(&mx[(long)g * 64 + c], __float_as_uint(v));
  if (c == 0) atomicAdd(&cnt[g], 1.0f);
}

// ---------------------------------------------------------------------------
// Classifier: pooled[16x128] @ wc1[128x64] (+relu) via WMMA, then 64->2 scalar
// Block: 128 threads = 4 waves; 16 graphs per block.
// ---------------------------------------------------------------------------
__global__ void classifier(const float* __restrict__ sum,
                           const unsigned* __restrict__ mx,
                           const float* __restrict__ cnt,
                           const float* __restrict__ wc1,  // [128,64]
                           const float* __restrict__ bc1,  // [64]
                           const float* __restrict__ wc2,  // [64,2]
                           const float* __restrict__ bc2,  // [2]
                           float* __restrict__ out,        // [G,2]
                           int G) {
  constexpr int SP = 132, SH = 68, SWP = 80;
  __shared__ float sP[16 * SP];
  __shared__ float sHg[16 * SH];
  __shared__ v2f sWp[64 * SWP];

  const int t = threadIdx.x;
  const int wave = t >> 5;
  const int lane = t & 31;
  const int half = lane >> 4;
  const int idx = lane & 15;
  const int n0 = wave * 16;
  const int g0 = blockIdx.x * 16;

  for (int i = t; i < 64 * 64; i += 128) {
    int pr = i >> 6, n = i & 63;
    v2f w;
    w.x = wc1[(2 * pr) * 64 + n];
    w.y = wc1[(2 * pr + 1) * 64 + n];
    sWp[pr * SWP + n] = w;
  }
  for (int i = t; i < 16 * 128; i += 128) {
    int r = i >> 7, c = i & 127;
    int g = g0 + r;
    float v = 0.0f;
    if (g < G) {
      if (c < 64)
        v = sum[(long)g * 64 + c] / fmaxf(cnt[g], 1.0f);   // mean pool
      else
        v = __uint_as_float(mx[(long)g * 64 + (c - 64)]);  // max pool
    }
    sP[r * SP + c] = v;
  }
  __syncthreads();

  v8f acc = {};
#pragma unroll
  for (int k = 0; k < 128; k += 4) {
    v2f a, b;
    a.x = sP[idx * SP + k + 2 * half];
    a.y = sP[idx * SP + k + 2 * half + 1];
    b = sWp[((k >> 1) + half) * SWP + n0 + idx];
    acc = wmma_f32(a, b, acc);
  }
  {
    float bias = bc1[n0 + idx];
#pragma unroll
    for (int v = 0; v < 8; ++v) {
      int row = v + 8 * half;
      sHg[row * SH + n0 + idx] = fmaxf(acc[v] + bias, 0.0f);
    }
  }
  __syncthreads();

  if (t < 32) {
    int row = t >> 1, col = t & 1;
    int g = g0 + row;
    if (g < G) {
      float s = bc2[col];
#pragma unroll 8
      for (int k = 0; k < 64; ++k) s += sHg[row * SH + k] * wc2[k * 2 + col];
      out[(long)g * 2 + col] = s;
    }
  }
}

// ---------------------------------------------------------------------------
extern "C" void kernel_launch(void* const* d_in, const int* in_sizes, int n_in,
                              void* d_out, int out_size, void* d_ws, size_t ws_size,
                              hipStream_t stream) {
  const float* x = (const float*)d_in[0];
  const int* ei = (const int*)d_in[1];
  const int* batch = (const int*)d_in[2];
  const int N = in_sizes[0] / 10;
  const int E = in_sizes[1] / 2;
  const int G = out_size / 2;
  const int* src = ei;
  const int* dst = ei + E;

  // conv params start at index 4 (conv1), 12 (conv2), 20 (conv3); cls at 28
  const float* p[32];
  for (int i = 4; i < n_in && i < 32; ++i) p[i] = (const float*)d_in[i];

  // workspace layout
  char* ws = (char*)d_ws;
  size_t off = 0;
  auto take = [&](size_t bytes) {
    char* r = ws + off;
    off += (bytes + 255) & ~(size_t)255;
    return r;
  };
  float* A = (float*)take((size_t)N * 10 * 4);   // layer-1 aggregated input
  float* B = (float*)take((size_t)N * 64 * 4);   // node features (ping)
  float* C = (float*)take((size_t)N * 64 * 4);   // node features (pong)
  float* sum = (float*)take((size_t)G * 64 * 4);
  unsigned* mx = (unsigned*)take((size_t)G * 64 * 4);
  float* cnt = (float*)take((size_t)G * 4);
  (void)ws_size;

  const int mlpBlocks = (N + 31) / 32;
  const int e64Blocks = (int)(((long)E * 64 + 255) / 256);
  const int poolBlocks = (int)(((long)N * 64 + 255) / 256);

  // ---- layer 1: agg = x + scatter(x); h1 = MLP1(agg) -> B ----
  hipMemcpyAsync(A, x, (size_t)N * 10 * 4, hipMemcpyDeviceToDevice, stream);
  scatter_add10<<<(E + 255) / 256, 256, 0, stream>>>(x, src, dst, A, E);
  gin_mlp<10, 12, 12><<<mlpBlocks, 256, 0, stream>>>(
      A, p[4], p[5], p[6], p[7], p[8], p[9], p[10], p[11], B, N);

  // ---- layer 2: C = B + scatter(B); h2 = MLP2(C) -> B ----
  hipMemcpyAsync(C, B, (size_t)N * 64 * 4, hipMemcpyDeviceToDevice, stream);
  scatter_add64<<<e64Blocks, 256, 0, stream>>>(B, src, dst, C, E);
  gin_mlp<64, 64, 68><<<mlpBlocks, 256, 0, stream>>>(
      C, p[12], p[13], p[14], p[15], p[16], p[17], p[18], p[19], B, N);

  // ---- layer 3 ----
  hipMemcpyAsync(C, B, (size_t)N * 64 * 4, hipMemcpyDeviceToDevice, stream);
  scatter_add64<<<e64Blocks, 256, 0, stream>>>(B, src, dst, C, E);
  gin_mlp<64, 64, 68><<<mlpBlocks, 256, 0, stream>>>(
      C, p[20], p[21], p[22], p[23], p[24], p[25], p[26], p[27], B, N);

  // ---- pooling (sum/max/count are contiguous -> one memset) ----
  hipMemsetAsync(sum, 0, (((size_t)G * 64 * 4 + 255) & ~(size_t)255) +
                             (((size_t)G * 64 * 4 + 255) & ~(size_t)255) +
                             (size_t)G * 4, stream);
  pool_accum<<<poolBlocks, 256, 0, stream>>>(B, batch, sum, mx, cnt, N);

  // ---- classifier ----
  classifier<<<(G + 15) / 16, 128, 0, stream>>>(
      sum, mx, cnt, p[28], p[29], p[30], p[31], (float*)d_out, G);
}